// GCNNet_23330262352264
// MI455X (gfx1250) — compile-verified
//
#include <hip/hip_runtime.h>

#define D    300
#define D4   75     // D / 4 (float4 chunks per row)
#define NT   19     // ceil(300/16) N-tiles (19*16 = 304, last tile masked)

typedef __attribute__((ext_vector_type(2))) float v2f;
typedef __attribute__((ext_vector_type(8))) float v8f;

// agg = src  (this realizes the self-loop edge: agg starts as h itself)
__global__ void copy_f4(float4* __restrict__ dstp, const float4* __restrict__ srcp, int n) {
    int i = blockIdx.x * blockDim.x + threadIdx.x;
    if (i < n) dstp[i] = srcp[i];
}

// agg[dst[e]] += h[src[e]]   one thread per (edge, float4 chunk)
__global__ void scatter_edges(const float* __restrict__ h,
                              const int* __restrict__ src,
                              const int* __restrict__ dst,
                              float* __restrict__ agg, int nE) {
    int idx = blockIdx.x * blockDim.x + threadIdx.x;
    int e = idx / D4;
    if (e >= nE) return;
    int c = idx - e * D4;
    int s = src[e];
    int d = dst[e];
    float4 v = reinterpret_cast<const float4*>(h + (size_t)s * D)[c];
    float* p = agg + (size_t)d * D + (size_t)c * 4;
    atomicAdd(p + 0, v.x);
    atomicAdd(p + 1, v.y);
    atomicAdd(p + 2, v.z);
    atomicAdd(p + 3, v.w);
}

// out = relu(A @ W^T + bias),  A:[N x 300], W:[300 x 300] row-major, out:[N x 300]
// One wave32 computes one 16x16 output tile via V_WMMA_F32_16X16X4_F32 (exact fp32).
__global__ void gemm_bias_relu(const float* __restrict__ A,
                               const float* __restrict__ W,
                               const float* __restrict__ bias,
                               float* __restrict__ out,
                               int nTilesM) {
    int wave = (int)((blockIdx.x * blockDim.x + threadIdx.x) >> 5);
    int lane = threadIdx.x & 31;
    int nTiles = nTilesM * NT;
    if (wave >= nTiles) return;

    int mt = wave / NT;
    int nt = wave - mt * NT;
    int m0 = mt * 16;
    int n0 = nt * 16;

    int half = lane >> 4;      // 0: K{0,1}, 1: K{2,3} per 16x4 f32 A/B layout
    int l    = lane & 15;

    int m = m0 + l;            // A row held by this lane (N is a multiple of 16)
    int j = n0 + l;            // W row == output column held by this lane
    bool jvalid = j < D;
    int jc = jvalid ? j : 0;   // clamp load address; garbage cols are never stored

    const float* arow = A + (size_t)m  * D + 2 * half;
    const float* wrow = W + (size_t)jc * D + 2 * half;

#if __has_builtin(__builtin_amdgcn_wmma_f32_16x16x4_f32)
    v8f acc0 = {};
    v8f acc1 = {};
    int k = 0;
    // 300 = 8*37 + 4 : dual accumulators to break the WMMA RAW chain
    for (; k + 8 <= D; k += 8) {
        v2f a0 = *(const v2f*)(arow + k);
        v2f b0 = *(const v2f*)(wrow + k);
        acc0 = __builtin_amdgcn_wmma_f32_16x16x4_f32(false, a0, false, b0,
                                                     (short)0, acc0, false, false);
        v2f a1 = *(const v2f*)(arow + k + 4);
        v2f b1 = *(const v2f*)(wrow + k + 4);
        acc1 = __builtin_amdgcn_wmma_f32_16x16x4_f32(false, a1, false, b1,
                                                     (short)0, acc1, false, false);
    }
    for (; k < D; k += 4) {
        v2f a0 = *(const v2f*)(arow + k);
        v2f b0 = *(const v2f*)(wrow + k);
        acc0 = __builtin_amdgcn_wmma_f32_16x16x4_f32(false, a0, false, b0,
                                                     (short)0, acc0, false, false);
    }
    if (jvalid) {
        float bj = bias[j];
        #pragma unroll
        for (int r = 0; r < 8; ++r) {
            // C/D layout: VGPR r, lanes 0-15 -> row m0+r, lanes 16-31 -> row m0+r+8
            float v = acc0[r] + acc1[r] + bj;
            v = v > 0.f ? v : 0.f;
            out[(size_t)(m0 + r + 8 * half) * D + j] = v;
        }
    }
#else
    // Scalar fallback (same output mapping) -- only used if the f32 WMMA builtin
    // is absent; histogram will show wmma==0 and we switch to the bf16 WMMA path.
    if (jvalid) {
        float bj = bias[j];
        for (int r = 0; r < 8; ++r) {
            int row = m0 + r + 8 * half;
            const float* ar = A + (size_t)row * D;
            const float* wr = W + (size_t)j * D;
            float s = bj;
            for (int k = 0; k < D; ++k) s += ar[k] * wr[k];
            out[(size_t)row * D + j] = s > 0.f ? s : 0.f;
        }
    }
#endif
}

extern "C" void kernel_launch(void* const* d_in, const int* in_sizes, int n_in,
                              void* d_out, int out_size, void* d_ws, size_t ws_size,
                              hipStream_t stream) {
    const float* features = (const float*)d_in[0];
    const int*   src      = (const int*)  d_in[1];
    const int*   dst      = (const int*)  d_in[2];
    const float* W1       = (const float*)d_in[3];
    const float* b1       = (const float*)d_in[4];
    const float* W2       = (const float*)d_in[5];
    const float* b2       = (const float*)d_in[6];

    const int N = in_sizes[0] / D;   // 10000
    const int E = in_sizes[1];       // 160000

    float* agg = (float*)d_ws;               // N*D floats (12 MB)
    float* h1  = agg + (size_t)N * D;        // N*D floats (12 MB)
    float* out = (float*)d_out;

    const int n4         = N * D4;
    const int copyBlocks = (n4 + 255) / 256;
    const int scatIdx    = E * D4;                 // 12M threads
    const int scatBlocks = (scatIdx + 255) / 256;
    const int nTilesM    = (N + 15) / 16;          // 625
    const int nTiles     = nTilesM * NT;           // 11875 waves
    const int gemmBlocks = (nTiles * 32 + 255) / 256;

    // ---- Layer 1: agg = features (self-loop); agg += scatter(features); h1 = relu(agg@W1^T+b1)
    copy_f4      <<<copyBlocks, 256, 0, stream>>>((float4*)agg, (const float4*)features, n4);
    scatter_edges<<<scatBlocks, 256, 0, stream>>>(features, src, dst, agg, E);
    gemm_bias_relu<<<gemmBlocks, 256, 0, stream>>>(agg, W1, b1, h1, nTilesM);

    // ---- Layer 2: agg = h1; agg += scatter(h1); out = relu(agg@W2^T+b2)
    copy_f4      <<<copyBlocks, 256, 0, stream>>>((float4*)agg, (const float4*)h1, n4);
    scatter_edges<<<scatBlocks, 256, 0, stream>>>(h1, src, dst, agg, E);
    gemm_bias_relu<<<gemmBlocks, 256, 0, stream>>>(agg, W2, b2, out, nTilesM);
}